// CBC_34033320854004
// MI455X (gfx1250) — compile-verified
//
#include <hip/hip_runtime.h>

typedef __attribute__((ext_vector_type(2))) float v2f;
typedef __attribute__((ext_vector_type(8))) float v8f;

#define DDIM 1024
#define KCOMP 5
#define CCLS 3

// One wave handles a 16-row tile of x. A-matrix (16x4 f32) layout per CDNA5 ISA:
//   lane l (l<16):  VGPR0 = A[M=l][K=0], VGPR1 = A[M=l][K=1]
//   lane l+16:      VGPR0 = A[M=l][K=2], VGPR1 = A[M=l][K=3]
// -> each lane loads a contiguous float2 at (row, d0 + 2*hi).
// B-matrix (4x16 f32) mirrors this with N across lanes; components live in LDS
// with a zero pad row so lanes with N >= 5 contribute exactly 0.
__global__ __launch_bounds__(256) void cbc_fused_kernel(
    const float* __restrict__ x,
    const float* __restrict__ comp,
    const float* __restrict__ rsn,
    float* __restrict__ out,
    int Brows) {
  __shared__ float ldsC[6][DDIM];       // rows 0..4 = components, row 5 = zeros
  __shared__ float dots[8][16][6];      // per-wave 16x5 dot tile (padded)
  __shared__ float x2s[8][16];          // per-wave row norms
  __shared__ float c2s[KCOMP];          // component norms

  const int tid  = threadIdx.x;
  const int wv   = tid >> 5;            // wave id 0..7
  const int lane = tid & 31;
  const int hi   = lane >> 4;           // 0: K={0,1}, 1: K={2,3}
  const int lo   = lane & 15;           // row (A) / column (B) index

  // ---- stage components into LDS (5*1024 floats = 1280 float4) + zero pad row
  {
    const float4* src = reinterpret_cast<const float4*>(comp);
    float4* dst = reinterpret_cast<float4*>(&ldsC[0][0]);
#pragma unroll
    for (int i = 0; i < 5; ++i) dst[tid + 256 * i] = src[tid + 256 * i];
#pragma unroll
    for (int i = 0; i < 4; ++i) ldsC[5][tid + 256 * i] = 0.0f;
  }
  __syncthreads();

  // ---- |c_k|^2 : waves 0..4 each reduce one component row
  if (wv < KCOMP) {
    float s = 0.0f;
#pragma unroll 4
    for (int i = 0; i < 32; ++i) {
      float v = ldsC[wv][lane + (i << 5)];
      s += v * v;
    }
#pragma unroll
    for (int off = 16; off > 0; off >>= 1) s += __shfl_xor(s, off, 32);
    if (lane == 0) c2s[wv] = s;
  }

  // ---- main WMMA loop: [16 x 1024] tile of x  dot  [1024 x 5] components
  const int rowBase = blockIdx.x * 128 + wv * 16;
  const int row     = rowBase + lo;
  const int rowL    = row < Brows ? row : (Brows - 1);
  const float* aPtr = x + (size_t)rowL * DDIM + 2 * hi;
  const float* bPtr = &ldsC[lo < KCOMP ? lo : 5][2 * hi];

  v8f acc0 = {};
  v8f acc1 = {};
  float sq = 0.0f;
#pragma unroll 4
  for (int d0 = 0; d0 < DDIM; d0 += 8) {
    v2f a0 = *reinterpret_cast<const v2f*>(aPtr + d0);
    v2f b0 = *reinterpret_cast<const v2f*>(bPtr + d0);
    acc0 = __builtin_amdgcn_wmma_f32_16x16x4_f32(false, a0, false, b0,
                                                 (short)0, acc0, false, false);
    sq += a0.x * a0.x + a0.y * a0.y;
    v2f a1 = *reinterpret_cast<const v2f*>(aPtr + d0 + 4);
    v2f b1 = *reinterpret_cast<const v2f*>(bPtr + d0 + 4);
    acc1 = __builtin_amdgcn_wmma_f32_16x16x4_f32(false, a1, false, b1,
                                                 (short)0, acc1, false, false);
    sq += a1.x * a1.x + a1.y * a1.y;
  }
  // lanes l and l+16 together cover all of row l -> combine halves
  sq += __shfl_xor(sq, 16, 32);

  // ---- stage dot tile + row norms to LDS (C/D layout: VGPR j -> M = j + 8*hi)
  if (lo < KCOMP) {
#pragma unroll
    for (int j = 0; j < 8; ++j)
      dots[wv][j + 8 * hi][lo] = acc0[j] + acc1[j];
  }
  if (hi == 0) x2s[wv][lo] = sq;
  __syncthreads();

  // ---- fused epilogue: lanes 0..15 of each wave own one row each
  if (hi == 0) {
    // reasoning-derived weights (clamped): pk = A, nk = (1-A)*B
    float w[CCLS][KCOMP];
    float bias[CCLS] = {0.0f, 0.0f, 0.0f};
    float den[CCLS]  = {0.0f, 0.0f, 0.0f};
#pragma unroll
    for (int k = 0; k < KCOMP; ++k) {
#pragma unroll
      for (int c = 0; c < CCLS; ++c) {
        float a = rsn[k * (CCLS * 2) + c * 2 + 0];
        float b = rsn[k * (CCLS * 2) + c * 2 + 1];
        a = fminf(fmaxf(a, 0.0f), 1.0f);
        b = fminf(fmaxf(b, 0.0f), 1.0f);
        float nk = (1.0f - a) * b;
        w[c][k] = a - nk;
        bias[c] += nk;
        den[c]  += a + nk;
      }
    }

    const float x2v = x2s[wv][lo];
    float num[CCLS] = {0.0f, 0.0f, 0.0f};
#pragma unroll
    for (int n = 0; n < KCOMP; ++n) {
      float d2 = x2v + c2s[n] - 2.0f * dots[wv][lo][n];
      d2 = fmaxf(d2, 0.0f);
      float sim = __expf(-0.5f * d2);
#pragma unroll
      for (int c = 0; c < CCLS; ++c) num[c] += sim * w[c][n];
    }

    const int orow = rowBase + lo;
    if (orow < Brows) {
#pragma unroll
      for (int c = 0; c < CCLS; ++c)
        out[orow * CCLS + c] = (num[c] + bias[c]) / den[c];
    }
  }
}

extern "C" void kernel_launch(void* const* d_in, const int* in_sizes, int n_in,
                              void* d_out, int out_size, void* d_ws, size_t ws_size,
                              hipStream_t stream) {
  (void)n_in; (void)out_size; (void)d_ws; (void)ws_size;
  const float* x    = (const float*)d_in[0];
  const float* comp = (const float*)d_in[1];
  const float* rsn  = (const float*)d_in[2];
  float* out        = (float*)d_out;

  const int Brows  = in_sizes[0] / DDIM;      // 32768
  const int blocks = (Brows + 127) / 128;     // 128 rows per block (8 waves x 16)

  hipLaunchKernelGGL(cbc_fused_kernel, dim3(blocks), dim3(256), 0, stream,
                     x, comp, rsn, out, Brows);
}